// apply_jpg_87909390615152
// MI455X (gfx1250) — compile-verified
//
#include <hip/hip_runtime.h>
#include <math.h>

typedef __attribute__((ext_vector_type(2))) float v2f;
typedef __attribute__((ext_vector_type(8))) float v8f;

// Orthonormal 8-point DCT-II matrix, precomputed (row 0 = 1/sqrt(8), rows k>0 = 0.5*cos((2j+1)k*pi/16))
__constant__ float c_D8[64] = {
    0.353553391f,  0.353553391f,  0.353553391f,  0.353553391f,  0.353553391f,  0.353553391f,  0.353553391f,  0.353553391f,
    0.490392640f,  0.415734806f,  0.277785117f,  0.097545161f, -0.097545161f, -0.277785117f, -0.415734806f, -0.490392640f,
    0.461939766f,  0.191341716f, -0.191341716f, -0.461939766f, -0.461939766f, -0.191341716f,  0.191341716f,  0.461939766f,
    0.415734806f, -0.097545161f, -0.490392640f, -0.277785117f,  0.277785117f,  0.490392640f,  0.097545161f, -0.415734806f,
    0.353553391f, -0.353553391f, -0.353553391f,  0.353553391f,  0.353553391f, -0.353553391f, -0.353553391f,  0.353553391f,
    0.277785117f, -0.490392640f,  0.097545161f,  0.415734806f, -0.415734806f, -0.097545161f,  0.490392640f, -0.277785117f,
    0.191341716f, -0.461939766f,  0.461939766f, -0.191341716f, -0.191341716f,  0.461939766f, -0.461939766f,  0.191341716f,
    0.097545161f, -0.277785117f,  0.415734806f, -0.490392640f,  0.490392640f, -0.415734806f,  0.277785117f, -0.097545161f};

__constant__ float c_LUM[64] = {
    16,11,10,16,24,40,51,61,  12,12,14,19,26,58,60,55,
    14,13,16,24,40,57,69,56,  14,17,22,29,51,87,80,62,
    18,22,37,56,68,109,103,77, 24,35,55,64,81,104,113,92,
    49,64,78,87,103,121,120,101, 72,92,95,98,112,100,103,99};
__constant__ float c_CHR[64] = {
    17,18,24,47,99,99,99,99,  18,21,26,66,99,99,99,99,
    24,26,56,99,99,99,99,99,  47,66,99,99,99,99,99,99,
    99,99,99,99,99,99,99,99,  99,99,99,99,99,99,99,99,
    99,99,99,99,99,99,99,99,  99,99,99,99,99,99,99,99};

__device__ __forceinline__ v8f wmma4(v2f a, v2f b, v8f c) {
    // D(16x16,f32) = A(16x4,f32) * B(4x16,f32) + C
    return __builtin_amdgcn_wmma_f32_16x16x4_f32(false, a, false, b, (short)0, c,
                                                 false, false);
}

#define LDS_STRIDE 17  // 16 + 1 pad -> conflict-free column reads

__global__ __launch_bounds__(256) void jpeg_roundtrip_kernel(
    const float* __restrict__ x, const int* __restrict__ qual,
    float* __restrict__ out)
{
    __shared__ float smem[8][16 * LDS_STRIDE];

    const int lane = threadIdx.x & 31;
    const int wave = threadIdx.x >> 5;
    const int m    = lane & 15;   // M (A,C/D) or N (B,C/D) index
    const int hi   = lane >> 4;   // lane-half select

    const int tile = blockIdx.x * 8 + wave;   // 32768 tiles total
    const int img  = tile >> 10;
    const int t    = tile & 1023;
    const int row0 = (t >> 5) << 4;
    const int col0 = (t & 31) << 4;

    float* ws = smem[wave];

    // ---- quality -> scale -> per-lane quant divisors (hoisted, computed once) ----
    int q = qual[0];
    q = q < 1 ? 1 : (q > 100 ? 100 : q);
    const float scale = (q < 50) ? (5000.0f / (float)q) : (200.0f - 2.0f * (float)q);
    const int qcol = m & 7;
    float qL[8], qC[8];
#pragma unroll
    for (int v = 0; v < 8; ++v) {
        float tl = floorf((c_LUM[v * 8 + qcol] * scale + 50.0f) / 100.0f);
        float tc = floorf((c_CHR[v * 8 + qcol] * scale + 50.0f) / 100.0f);
        qL[v] = fminf(fmaxf(tl, 1.0f), 255.0f);
        qC[v] = fminf(fmaxf(tc, 1.0f), 255.0f);
    }

    // ---- constant DCT fragments from the precomputed table ----
    // dA[s]: A-frag of Dbig == B-frag of Dbig^T   (Dbig[m][kk])
    // dB[s]: A-frag of Dbig^T == B-frag of Dbig   (Dbig[kk][m])
    v2f dA[4], dB[4];
#pragma unroll
    for (int s = 0; s < 4; ++s)
#pragma unroll
        for (int j = 0; j < 2; ++j) {
            int kk = 4 * s + j + 2 * hi;
            bool diag = (m >> 3) == (kk >> 3);
            dA[s][j] = diag ? c_D8[(m & 7) * 8 + (kk & 7)] : 0.0f;
            dB[s][j] = diag ? c_D8[(kk & 7) * 8 + (m & 7)] : 0.0f;
        }

    // ---- load RGB tile in B-fragment layout; convert to YCC-128 ----
    const size_t plane = (size_t)512 * 512;
    const float* base  = x + (size_t)img * 3 * plane;
    v2f xc[3][4];
#pragma unroll
    for (int s = 0; s < 4; ++s)
#pragma unroll
        for (int j = 0; j < 2; ++j) {
            int r = 4 * s + j + 2 * hi;
            size_t off = (size_t)(row0 + r) * 512 + (size_t)(col0 + m);
            float rr = base[0 * plane + off];
            float gg = base[1 * plane + off];
            float bb = base[2 * plane + off];
            rr = rintf(fminf(fmaxf(rr, 0.0f), 1.0f) * 255.0f);
            gg = rintf(fminf(fmaxf(gg, 0.0f), 1.0f) * 255.0f);
            bb = rintf(fminf(fmaxf(bb, 0.0f), 1.0f) * 255.0f);
            xc[0][s][j] =  0.299f    * rr + 0.587f    * gg + 0.114f * bb - 128.0f;
            xc[1][s][j] = -0.168736f * rr - 0.331264f * gg + 0.5f   * bb;
            xc[2][s][j] =  0.5f      * rr - 0.418688f * gg - 0.081312f * bb;
        }

    float rec[3][8];

#pragma unroll
    for (int ch = 0; ch < 3; ++ch) {
        // ---- Y1 = Dbig * X  (X as B-fragments) ----
        v8f acc = {};
#pragma unroll
        for (int s = 0; s < 4; ++s) acc = wmma4(dA[s], xc[ch][s], acc);

        // C/D layout -> LDS row-major
#pragma unroll
        for (int v = 0; v < 8; ++v)
            ws[(v + 8 * hi) * LDS_STRIDE + m] = acc[v];
        asm volatile("s_wait_dscnt 0" ::: "memory");
        v2f y1a[4];
#pragma unroll
        for (int s = 0; s < 4; ++s)
#pragma unroll
            for (int j = 0; j < 2; ++j)
                y1a[s][j] = ws[m * LDS_STRIDE + 4 * s + j + 2 * hi];
        asm volatile("s_wait_dscnt 0" ::: "memory");

        // ---- coef = Y1 * Dbig^T ----
        v8f cf = {};
#pragma unroll
        for (int s = 0; s < 4; ++s) cf = wmma4(y1a[s], dA[s], cf);

        // ---- quantize: coef = round(coef/q)*q (exact fp32 division) ----
        const float* qv = (ch == 0) ? qL : qC;
#pragma unroll
        for (int v = 0; v < 8; ++v)
            cf[v] = rintf(cf[v] / qv[v]) * qv[v];

        // C/D -> LDS, reload as B-fragments of coef
#pragma unroll
        for (int v = 0; v < 8; ++v)
            ws[(v + 8 * hi) * LDS_STRIDE + m] = cf[v];
        asm volatile("s_wait_dscnt 0" ::: "memory");
        v2f cfb[4];
#pragma unroll
        for (int s = 0; s < 4; ++s)
#pragma unroll
            for (int j = 0; j < 2; ++j)
                cfb[s][j] = ws[(4 * s + j + 2 * hi) * LDS_STRIDE + m];
        asm volatile("s_wait_dscnt 0" ::: "memory");

        // ---- T1 = Dbig^T * coef ----
        v8f t1 = {};
#pragma unroll
        for (int s = 0; s < 4; ++s) t1 = wmma4(dB[s], cfb[s], t1);

        // C/D -> LDS, reload as A-fragments of T1
#pragma unroll
        for (int v = 0; v < 8; ++v)
            ws[(v + 8 * hi) * LDS_STRIDE + m] = t1[v];
        asm volatile("s_wait_dscnt 0" ::: "memory");
        v2f t1a[4];
#pragma unroll
        for (int s = 0; s < 4; ++s)
#pragma unroll
            for (int j = 0; j < 2; ++j)
                t1a[s][j] = ws[m * LDS_STRIDE + 4 * s + j + 2 * hi];
        asm volatile("s_wait_dscnt 0" ::: "memory");

        // ---- rec = T1 * Dbig ----
        v8f rc = {};
#pragma unroll
        for (int s = 0; s < 4; ++s) rc = wmma4(t1a[s], dB[s], rc);

#pragma unroll
        for (int v = 0; v < 8; ++v) rec[ch][v] = rc[v];
    }

    // ---- YCC -> RGB, clamp/round, store (C/D layout) ----
    float* obase = out + (size_t)img * 3 * plane;
#pragma unroll
    for (int v = 0; v < 8; ++v) {
        size_t off = (size_t)(row0 + v + 8 * hi) * 512 + (size_t)(col0 + m);
        float yy  = rec[0][v] + 128.0f;
        float cbb = rec[1][v];
        float crr = rec[2][v];
        float rr = yy + 1.402f * crr;
        float gg = yy - 0.344136f * cbb - 0.714136f * crr;
        float bb = yy + 1.772f * cbb;
        obase[0 * plane + off] = rintf(fminf(fmaxf(rr, 0.0f), 255.0f)) * (1.0f / 255.0f);
        obase[1 * plane + off] = rintf(fminf(fmaxf(gg, 0.0f), 255.0f)) * (1.0f / 255.0f);
        obase[2 * plane + off] = rintf(fminf(fmaxf(bb, 0.0f), 255.0f)) * (1.0f / 255.0f);
    }
}

extern "C" void kernel_launch(void* const* d_in, const int* in_sizes, int n_in,
                              void* d_out, int out_size, void* d_ws, size_t ws_size,
                              hipStream_t stream) {
    const float* x    = (const float*)d_in[0];
    const int*   qual = (const int*)d_in[1];
    float*       out  = (float*)d_out;
    // 32 images * (512/16)^2 tiles = 32768 tiles; 8 waves (tiles) per 256-thread block
    dim3 grid(4096), block(256);
    hipLaunchKernelGGL(jpeg_roundtrip_kernel, grid, block, 0, stream, x, qual, out);
}